// CustomSVRActivationLayer_39934605918594
// MI455X (gfx1250) — compile-verified
//
#include <hip/hip_runtime.h>
#include <hip/hip_bf16.h>

typedef __attribute__((ext_vector_type(2))) float v2f;
typedef __attribute__((ext_vector_type(8))) float v8f;

#define B_ROWS  4096
#define F_COLS  8192
#define NCLS    1000
#define SPLITS  32
#define K_PER_WAVE   (F_COLS / SPLITS)     // 256 floats of each row per wave
#define KTILES       (K_PER_WAVE / 4)      // 64 WMMA K-tiles per wave

// ---------------------------------------------------------------------------
// Kernel 1: per-row sum and sum-of-squares partials via V_WMMA_F32_16X16X4_F32
// with an all-ones B matrix (D[m,n] += sum_k A[m,k]).
// A layout (ISA 7.12.2, 32-bit A 16x4): lanes 0-15 hold (K=0,K=1) of row M=lane,
// lanes 16-31 hold (K=2,K=3) of row M=lane-16  -> one float2 load per lane.
// C/D layout: lane 0 holds D[0..7][0] in its 8 VGPRs, lane 16 holds D[8..15][0].
// ---------------------------------------------------------------------------
__global__ __launch_bounds__(32)
void rowstats_wmma(const float* __restrict__ in,
                   float* __restrict__ ws_sum,
                   float* __restrict__ ws_sq) {
    const int lane    = threadIdx.x;        // 0..31 (wave32)
    const int rowTile = blockIdx.x;         // 0..255  (16 rows each)
    const int split   = blockIdx.y;         // 0..SPLITS-1 (K range)
    const int m       = lane & 15;
    const int khalf   = lane >> 4;          // 0: K=0,1   1: K=2,3

    const int row = rowTile * 16 + m;
    const float* p = in + (size_t)row * F_COLS + (size_t)split * K_PER_WAVE + khalf * 2;

    const v2f ones = {1.0f, 1.0f};
    v8f acc_s = {};   // row sums
    v8f acc_q = {};   // row sums of squares

    #pragma unroll 4
    for (int t = 0; t < KTILES; ++t) {
        v2f a  = *(const v2f*)(p + 4 * t);
        v2f a2 = a * a;
        acc_s = __builtin_amdgcn_wmma_f32_16x16x4_f32(
            false, a,  false, ones, (short)0, acc_s, false, false);
        acc_q = __builtin_amdgcn_wmma_f32_16x16x4_f32(
            false, a2, false, ones, (short)0, acc_q, false, false);
    }

    // Every column of D is identical; lane 0 owns rows tile+0..7 (VGPR i = row i),
    // lane 16 owns rows tile+8..15.
    if (lane == 0 || lane == 16) {
        const int rbase = rowTile * 16 + khalf * 8;
        float* os = ws_sum + (size_t)split * B_ROWS + rbase;
        float* oq = ws_sq  + (size_t)split * B_ROWS + rbase;
        #pragma unroll
        for (int i = 0; i < 8; ++i) { os[i] = acc_s[i]; oq[i] = acc_q[i]; }
    }
}

// ---------------------------------------------------------------------------
// Kernel 2: fold the SPLITS partials per row (deterministic order), gather
// y[b] = wm.flat[label[b]], and produce per-block partial sums of y.
// ---------------------------------------------------------------------------
__global__ __launch_bounds__(256)
void finalize_stats(const float* __restrict__ ws_sum,
                    const float* __restrict__ ws_sq,
                    const float* __restrict__ wm,
                    const int*   __restrict__ label,
                    float* __restrict__ sx,
                    float* __restrict__ sq,
                    float* __restrict__ yv,
                    float* __restrict__ ypart) {
    __shared__ float sh[256];
    const int b = blockIdx.x * 256 + threadIdx.x;   // 16 blocks x 256 = 4096

    float s = 0.0f, q = 0.0f;
    #pragma unroll 8
    for (int k = 0; k < SPLITS; ++k) {
        s += ws_sum[(size_t)k * B_ROWS + b];
        q += ws_sq [(size_t)k * B_ROWS + b];
    }
    sx[b] = s;
    sq[b] = q;

    const int lab = label[b];                 // 0..999
    const float y = wm[lab];                  // wm[0, lab, 0] == flat[lab]
    yv[b] = y;

    sh[threadIdx.x] = y;
    __syncthreads();
    #pragma unroll
    for (int off = 128; off > 0; off >>= 1) {
        if (threadIdx.x < off) sh[threadIdx.x] += sh[threadIdx.x + off];
        __syncthreads();
    }
    if (threadIdx.x == 0) ypart[blockIdx.x] = sh[0];
}

// ---------------------------------------------------------------------------
// Kernel 3: fold the 16 block partials -> y_avg (single deterministic chain).
// ---------------------------------------------------------------------------
__global__ void reduce_y(const float* __restrict__ ypart, float* __restrict__ yavg) {
    if (threadIdx.x == 0) {
        float s = 0.0f;
        #pragma unroll
        for (int i = 0; i < 16; ++i) s += ypart[i];
        *yavg = s / (float)B_ROWS;
    }
}

// ---------------------------------------------------------------------------
// Kernel 4: per-row closed-form scalar, broadcast across the 1000 columns.
// out[b, r] = upd[b].  Coalesced 1000-wide stores, 15.6 MiB total.
// ---------------------------------------------------------------------------
__global__ __launch_bounds__(256)
void write_out(const float* __restrict__ sx,
               const float* __restrict__ sq,
               const float* __restrict__ yv,
               const float* __restrict__ yavg,
               float* __restrict__ out) {
    const int b = blockIdx.x;
    const float s  = sx[b];
    const float q  = sq[b];
    const float y  = yv[b];
    const float ya = *yavg;

    const float xa   = s / (float)B_ROWS;                       // n_new = B
    const float Sx   = q - 2.0f * xa * s + (float)F_COLS * xa * xa;
    const float Sxy  = (y - ya) * (s - (float)F_COLS * xa);
    const float beta = Sxy / Sx;
    const float alph = ya - beta * xa;
    const float upd  = beta * s + alph;

    float* o = out + (size_t)b * NCLS;
    for (int r = threadIdx.x; r < NCLS; r += 256) o[r] = upd;
}

// ---------------------------------------------------------------------------
extern "C" void kernel_launch(void* const* d_in, const int* in_sizes, int n_in,
                              void* d_out, int out_size, void* d_ws, size_t ws_size,
                              hipStream_t stream) {
    const float* inputs = (const float*)d_in[0];   // (4096, 8192) f32
    const float* wm     = (const float*)d_in[1];   // (1000, 1000, 1) f32
    const int*   label  = (const int*)  d_in[2];   // (4096,) int
    float*       out    = (float*)d_out;           // (4096, 1000) f32

    // Workspace carve-up (floats).
    float* ws     = (float*)d_ws;
    float* ws_sum = ws;                                   // SPLITS * B
    float* ws_sq  = ws_sum + (size_t)SPLITS * B_ROWS;     // SPLITS * B
    float* sx     = ws_sq  + (size_t)SPLITS * B_ROWS;     // B
    float* sq     = sx + B_ROWS;                          // B
    float* yv     = sq + B_ROWS;                          // B
    float* ypart  = yv + B_ROWS;                          // 16
    float* yavg   = ypart + 16;                           // 1

    rowstats_wmma<<<dim3(B_ROWS / 16, SPLITS), 32, 0, stream>>>(inputs, ws_sum, ws_sq);
    finalize_stats<<<B_ROWS / 256, 256, 0, stream>>>(ws_sum, ws_sq, wm, label,
                                                     sx, sq, yv, ypart);
    reduce_y<<<1, 32, 0, stream>>>(ypart, yavg);
    write_out<<<B_ROWS, 256, 0, stream>>>(sx, sq, yv, yavg, out);
}